// MLLoss_17025250362030
// MI455X (gfx1250) — compile-verified
//
#include <hip/hip_runtime.h>
#include <math.h>

// ---------------------------------------------------------------------------
// CTC log-semiring forward DP for MI455X (gfx1250, wave32).
//   B=32 sequences, T=1024 frames, V=1024 vocab, L=256 labels, S=513 states.
// One workgroup per sequence. alpha ping-pong in LDS (self-alpha in register).
// Emission rows (4KB each) streamed into an 8-deep LDS ring buffer by the
// CDNA5 Tensor Data Mover, keeping up to 7 DMAs in flight to hide HBM latency
// behind the serial T-chain. TDM completions are in-order per wave, so
// s_wait_tensorcnt(DEPTH-2) releases exactly the oldest row.
// Kernel symbol name encodes the selected staging mode so the disasm snippet
// proves which path compiled (ctc_fwd_tdm / ctc_fwd_asynclds / ctc_fwd_sync).
// ---------------------------------------------------------------------------

#define B_  32
#define T_  1024
#define V_  1024
#define L_  256
#define S_  (2 * L_ + 1)      // 513 extended (blank-interleaved) states
#define NEGF (-1e30f)
#define NTHREADS 544          // 17 wave32 waves, covers S_=513 states
#define DEPTH 8               // ring-buffer depth (power of two)

// ---- staging-mode selection (compile-only environment; toolchain may vary) --
#if __has_builtin(__builtin_amdgcn_tensor_load_to_lds) && __has_builtin(__builtin_amdgcn_s_wait_tensorcnt)
  #define STAGE_MODE 2        // CDNA5 TDM async tensor DMA, TENSORcnt
  #define FWD_KERNEL ctc_fwd_tdm
#elif __has_builtin(__builtin_amdgcn_global_load_async_to_lds_b128) && __has_builtin(__builtin_amdgcn_s_wait_asynccnt)
  #define STAGE_MODE 1        // async global->LDS, ASYNCcnt
  #define FWD_KERNEL ctc_fwd_asynclds
#else
  #define STAGE_MODE 0        // synchronous fallback
  #define FWD_KERNEL ctc_fwd_sync
#endif

typedef __attribute__((address_space(3))) void       lds_void;
typedef __attribute__((address_space(1))) const void gbl_cvoid;

__device__ __forceinline__ float lae2(float a, float b) {
  float m = fmaxf(a, b);
  float d = fabsf(a - b);
  return m + __logf(1.0f + __expf(-d));
}

#if STAGE_MODE == 2
typedef unsigned int u32x4 __attribute__((ext_vector_type(4)));
typedef int          i32x4 __attribute__((ext_vector_type(4)));
typedef int          i32x8 __attribute__((ext_vector_type(8)));

// Issue one TDM descriptor: copy V_ contiguous f32 (one emission row) -> LDS.
// D# layout per CDNA5 ISA.
__device__ __forceinline__ void tdm_issue_row(const float* gsrc, void* ldsDst) {
  unsigned lds_off = (unsigned)(unsigned long long)(lds_void*)ldsDst;
  unsigned long long ga = (unsigned long long)(const void*)gsrc;

  u32x4 g0;
  g0[0] = 1u;                                      // count=1, user descriptor
  g0[1] = lds_off;                                 // lds_addr (bytes)
  g0[2] = (unsigned)(ga & 0xFFFFFFFFull);          // global_addr[31:0]
  g0[3] = (unsigned)((ga >> 32) & 0x1FFFFFFull)    // global_addr[56:32]
        | (2u << 30);                              // type = 2 ("image")

  i32x8 g1;
  g1[0] = (int)(2u << 16);                         // data_size = 4 bytes
  g1[1] = (int)(((unsigned)V_ & 0xFFFFu) << 16);   // tensor_dim0[15:0]
  g1[2] = (int)((((unsigned)V_ >> 16) & 0xFFFFu)   // tensor_dim0[31:16]
        | (1u << 16));                             // tensor_dim1 = 1
  g1[3] = (int)(((unsigned)V_ & 0xFFFFu) << 16);   // tile_dim0 = V_
  g1[4] = 0;                                       // tile_dim1=0, tile_dim2=0
  g1[5] = (int)V_;                                 // tensor_dim0_stride
  g1[6] = 0;
  g1[7] = 0;

  i32x4 gz = {0, 0, 0, 0};
#if defined(__clang_major__) && (__clang_major__ >= 23)
  i32x8 gz8 = {0, 0, 0, 0, 0, 0, 0, 0};
  __builtin_amdgcn_tensor_load_to_lds(g0, g1, gz, gz, gz8, 0);
#else
  __builtin_amdgcn_tensor_load_to_lds(g0, g1, gz, gz, 0);
#endif
}
#endif  // STAGE_MODE == 2

__device__ __forceinline__ void stage_issue(const float* gsrc, float* ldsDst, int tid) {
#if STAGE_MODE == 2
  if (tid < 32) tdm_issue_row(gsrc, (void*)ldsDst);   // one DMA per workgroup (wave 0)
#elif STAGE_MODE == 1
  if (tid < 256)
    __builtin_amdgcn_global_load_async_to_lds_b128(
        (gbl_cvoid*)(gsrc + tid * 4), (lds_void*)(ldsDst + tid * 4), 0, 0);
#else
  if (tid < 256) {
    float4 v = *(const float4*)(gsrc + tid * 4);
    *(float4*)(ldsDst + tid * 4) = v;               // visibility via __syncthreads
  }
#endif
}

// Wait until the OLDEST in-flight row has landed. In steady state DEPTH-1
// rows are outstanding -> wait cnt<=DEPTH-2. In the drain tail, conservatively
// wait for everything (immediates must be compile-time constants for SOPP).
__device__ __forceinline__ void stage_wait(int tid, bool steady) {
#if STAGE_MODE == 2
  if (tid < 32) {
    if (steady) __builtin_amdgcn_s_wait_tensorcnt(DEPTH - 2);
    else        __builtin_amdgcn_s_wait_tensorcnt(0);
  }
#elif STAGE_MODE == 1
  if (tid < 256) {
    if (steady) __builtin_amdgcn_s_wait_asynccnt(DEPTH - 2);
    else        __builtin_amdgcn_s_wait_asynccnt(0);
  }
#else
  (void)tid; (void)steady;
#endif
}

extern "C" __global__ __launch_bounds__(NTHREADS)
void FWD_KERNEL(const float* __restrict__ log_probs,
                const int*   __restrict__ targets,
                const int*   __restrict__ input_lengths,
                const int*   __restrict__ target_lengths,
                float*       __restrict__ loglik) {
  const int b   = blockIdx.x;
  const int tid = threadIdx.x;

  __shared__ float          s_alpha[2][S_ + 2];        // +2 front pad = NEGF
  __shared__ __align__(16) float s_row[DEPTH][V_];     // emission-row ring buffer

  const float* base = log_probs + (size_t)b * T_ * V_;
  const int*   tgt  = targets + b * L_;

  // ---- time-invariant per-state constants, kept in registers ---------------
  int  myExt  = 0;        // ext[s]: even s -> blank(0), odd s -> label
  bool mySkip = false;    // s>=2 && ext[s]!=0 && ext[s]!=ext[s-2]
  float aSelf = NEGF;     // alpha[s], state s == tid
  if (tid < S_) {
    if (tid & 1) {
      int l = tid >> 1;                   // (tid-1)/2 for odd tid
      myExt = tgt[l];
      if (tid >= 3) mySkip = (myExt != tgt[l - 1]);
    }
    if (tid == 0) aSelf = base[0];        // alpha0[0] = lp(t=0, blank)
    if (tid == 1) aSelf = base[myExt];    // alpha0[1] = lp(t=0, y0)
    s_alpha[0][tid + 2] = aSelf;
    s_alpha[1][tid + 2] = NEGF;
  }
  if (tid < 2) { s_alpha[0][tid] = NEGF; s_alpha[1][tid] = NEGF; }

  int in_len = input_lengths[b];
  if (in_len > T_) in_len = T_;
  if (in_len < 1)  in_len = 1;
  const int tEnd = in_len;   // alpha frozen for t >= input_len -> stop early

  // ---- prime the DMA pipeline: rows 1 .. min(DEPTH-1, tEnd-1) --------------
  {
    int P = tEnd - 1; if (P > DEPTH - 1) P = DEPTH - 1;
    for (int k = 1; k <= P; ++k)
      stage_issue(base + (size_t)k * V_, &s_row[k & (DEPTH - 1)][0], tid);
  }
  __syncthreads();

  for (int t = 1; t < tEnd; ++t) {
    const bool steady = ((tEnd - 1) - t) >= (DEPTH - 2);
    stage_wait(tid, steady);          // row t landed (TDM in-order per wave)
    __syncthreads();                  // publish row t + alpha(t-1) to all waves

    // refill the ring: slot (t-1)&7 was consumed last step
    if (t + DEPTH - 1 < tEnd)
      stage_issue(base + (size_t)(t + DEPTH - 1) * V_,
                  &s_row[(t + DEPTH - 1) & (DEPTH - 1)][0], tid);
    if (((tid >> 5) == 1) && (t + DEPTH + 1 < tEnd)) {   // wave 1: L2 prefetch
      const char* pf = (const char*)(base + (size_t)(t + DEPTH + 1) * V_);
      __builtin_prefetch(pf + (size_t)(tid & 31) * 128, 0, 0);
    }

    if (tid < S_) {
      const float* aOld = s_alpha[(t + 1) & 1];   // buffer written at step t-1
      float a1 = aOld[tid + 1];
      float a2 = mySkip ? aOld[tid] : NEGF;
      // 3-way logsumexp: max3 + 3 independent exps + 1 log (short dep chain)
      float m   = fmaxf(fmaxf(aSelf, a1), a2);
      float sum = __expf(aSelf - m) + __expf(a1 - m) + __expf(a2 - m);
      aSelf = m + __logf(sum) + s_row[t & (DEPTH - 1)][myExt];
      s_alpha[t & 1][tid + 2] = aSelf;
    }
  }
  __syncthreads();

  if (tid == 0) {
    int tl = target_lengths[b];
    if (tl < 1)  tl = 1;
    if (tl > L_) tl = L_;
    const float* aF = s_alpha[(tEnd - 1) & 1];
    loglik[b] = lae2(aF[2 * tl + 2], aF[2 * tl + 1]);
  }
}

// Stable descending sort by input length (rank formula), negate, mask.
extern "C" __global__ void ctc_finalize(const float* __restrict__ loglik,
                                        const int*   __restrict__ input_lengths,
                                        float* __restrict__ out, int out_size) {
  int tid = threadIdx.x;
  for (int i = tid; i < out_size; i += blockDim.x) out[i] = 0.0f;
  __syncthreads();
  if (tid < B_) {
    int myLen = input_lengths[tid];
    int rank = 0;
    for (int j = 0; j < B_; ++j) {
      int lj = input_lengths[j];
      rank += (lj > myLen) || (lj == myLen && j < tid);
    }
    float loss = -loglik[tid];
    int valid = isfinite(loss) && (fabsf(loss) < 1e29f);
    if (rank < out_size)      out[rank]      = valid ? loss : 0.0f;
    if (B_ + rank < out_size) out[B_ + rank] = valid ? 1.0f : 0.0f;
  }
}

extern "C" void kernel_launch(void* const* d_in, const int* in_sizes, int n_in,
                              void* d_out, int out_size, void* d_ws, size_t ws_size,
                              hipStream_t stream) {
  (void)in_sizes; (void)n_in; (void)ws_size;
  const float* log_probs      = (const float*)d_in[0];
  const int*   targets        = (const int*)d_in[1];
  const int*   input_lengths  = (const int*)d_in[2];
  const int*   target_lengths = (const int*)d_in[3];
  float* loglik = (float*)d_ws;   // 32 floats of scratch

  FWD_KERNEL<<<B_, NTHREADS, 0, stream>>>(
      log_probs, targets, input_lengths, target_lengths, loglik);
  ctc_finalize<<<1, 64, 0, stream>>>(
      loglik, input_lengths, (float*)d_out, out_size);
}